// DummyModule_11879879542396
// MI455X (gfx1250) — compile-verified
//
#include <hip/hip_runtime.h>

typedef __attribute__((ext_vector_type(2))) float v2f;
typedef __attribute__((ext_vector_type(8))) float v8f;

#define NPTS    4096
#define NSAMPLE 5

__device__ __forceinline__ unsigned bmux(unsigned a, unsigned b, unsigned cond01) {
    // branchless select: cond01 in {0,1}; returns cond01 ? b : a (pure VALU)
    return a ^ ((a ^ b) & (0u - cond01));
}

__global__ __launch_bounds__(256) void ball_query_wmma(const float* __restrict__ x,
                                                       int* __restrict__ out) {
    __shared__ float4 pts[NPTS];            // 64 KB: (x, y, z, |p|^2)

    const float r2   = 3.4f * 3.4f;
    const int   b    = blockIdx.x >> 5;     // 32 blocks per batch
    const int   blkB = blockIdx.x & 31;
    const int   tid  = threadIdx.x;
    const int   lane = tid & 31;            // wave32
    const int   wave = tid >> 5;            // 8 waves per block
    const bool  hi   = lane >= 16;          // upper half-wave carries K=2,3

    // ---- Stage whole batch into LDS (points + squared norms) ----
    const float* xb = x + (size_t)b * NPTS * 3;
    for (int i = tid; i < NPTS; i += 256) {
        float px = xb[i * 3 + 0];
        float py = xb[i * 3 + 1];
        float pz = xb[i * 3 + 2];
        pts[i] = make_float4(px, py, pz, px * px + py * py + pz * pz);
    }
    __syncthreads();

    const int qbase = (blkB * 8 + wave) * 16;   // 16 query rows per wave

    // ---- A operand: row m = (-2x, -2y, -2z, |x_m|^2), 16x4 f32 layout ----
    v2f a;
    {
        float4 p = pts[qbase + (lane & 15)];
        a.x = hi ? (-2.0f * p.z) : (-2.0f * p.x);   // K=2 : K=0
        a.y = hi ? p.w           : (-2.0f * p.y);   // K=3 : K=1
    }

    int cnt = hi ? NSAMPLE : 0;             // upper lanes never take slots
    int s0 = 0, s1 = 0, s2 = 0, s3 = 0, s4 = 0;

    for (int kt = 0; kt < NPTS / 16; ++kt) {
        const int kbase = kt * 16;

        // ---- B operand: col n = (x, y, z, 1); C = |x_n|^2 (branchless) ----
        float4 p = pts[kbase + (lane & 15)];        // one ds_load_b128
        v2f bm;
        bm.x = hi ? p.z  : p.x;                     // K=2 : K=0
        bm.y = hi ? 1.0f : p.y;                     // K=3 : K=1
        v8f c;
#pragma unroll
        for (int i = 0; i < 8; ++i) c[i] = p.w;

        // d2[m][n] tile in one matrix op
        v8f d = __builtin_amdgcn_wmma_f32_16x16x4_f32(
            false, a, false, bm, (short)0, c, false, false);

        // ---- 8 ballots -> 16-bit hit masks for the 16 rows ----
        unsigned bal0 = (unsigned)__ballot(d[0] < r2);
        unsigned bal1 = (unsigned)__ballot(d[1] < r2);
        unsigned bal2 = (unsigned)__ballot(d[2] < r2);
        unsigned bal3 = (unsigned)__ballot(d[3] < r2);
        unsigned bal4 = (unsigned)__ballot(d[4] < r2);
        unsigned bal5 = (unsigned)__ballot(d[5] < r2);
        unsigned bal6 = (unsigned)__ballot(d[6] < r2);
        unsigned bal7 = (unsigned)__ballot(d[7] < r2);

        // branchless 8:1 mux of the row's ballot (row = lane&15, vgpr = lane&7)
        const unsigned r0 = (unsigned)(lane & 1);
        const unsigned r1 = (unsigned)((lane >> 1) & 1);
        const unsigned r2b = (unsigned)((lane >> 2) & 1);
        unsigned s01 = bmux(bal0, bal1, r0);
        unsigned s23 = bmux(bal2, bal3, r0);
        unsigned s45 = bmux(bal4, bal5, r0);
        unsigned s67 = bmux(bal6, bal7, r0);
        unsigned s03 = bmux(s01, s23, r1);
        unsigned s47 = bmux(s45, s67, r1);
        unsigned sel = bmux(s03, s47, r2b);
        // rows 0-7 live in bits [15:0], rows 8-15 in bits [31:16]
        unsigned rm = (sel >> ((lane & 8) << 1)) & 0xFFFFu;

        // ---- branchless ordered take: consume lowest set bit at slot j iff cnt==j ----
#pragma unroll
        for (int j = 0; j < NSAMPLE; ++j) {
            int  idx  = kbase + (int)__builtin_ctz(rm | 0x10000u);
            bool take = (cnt == j) & (rm != 0u);
            if (j == 0) s0 = take ? idx : s0;
            if (j == 1) s1 = take ? idx : s1;
            if (j == 2) s2 = take ? idx : s2;
            if (j == 3) s3 = take ? idx : s3;
            if (j == 4) s4 = take ? idx : s4;
            cnt += (int)take;
            unsigned cleared = rm & (rm - 1u);
            rm = take ? cleared : rm;
        }

        // uniform early exit once every row in the wave has 5 hits
        if (__all(cnt >= NSAMPLE)) break;
    }

    if (lane < 16) {
        if (cnt < 2) s1 = s0;   // pad with first hit (s0==0 if none, matching ref)
        if (cnt < 3) s2 = s0;
        if (cnt < 4) s3 = s0;
        if (cnt < 5) s4 = s0;
        const int q = qbase + lane;
        size_t o = ((size_t)b * NPTS + q) * NSAMPLE;
        out[o + 0] = s0;
        out[o + 1] = s1;
        out[o + 2] = s2;
        out[o + 3] = s3;
        out[o + 4] = s4;
    }
}

extern "C" void kernel_launch(void* const* d_in, const int* in_sizes, int n_in,
                              void* d_out, int out_size, void* d_ws, size_t ws_size,
                              hipStream_t stream) {
    const float* x = (const float*)d_in[0];
    int* out = (int*)d_out;                 // reference output dtype is int32
    (void)in_sizes; (void)n_in; (void)out_size; (void)d_ws; (void)ws_size;
    // 8 batches * 32 blocks; 256 threads = 8 wave32 waves, 16 queries per wave
    ball_query_wmma<<<dim3(256), dim3(256), 0, stream>>>(x, out);
}